// GCNConv_80367428042848
// MI455X (gfx1250) — compile-verified
//
#include <hip/hip_runtime.h>

typedef __attribute__((ext_vector_type(2))) float v2f;
typedef __attribute__((ext_vector_type(8))) float v8f;

#define N_NODES 50000
#define N_EDGES 625000
#define CH 128
#define ROW_BLOCKS (N_NODES / 16)   // 3125 exactly

// ---------------------------------------------------------------------------
// Kernel 1: zero the aggregate buffer (d_out, reused in-place) and deg (ws).
// ---------------------------------------------------------------------------
__global__ void k_zero(float* __restrict__ agg, float* __restrict__ deg) {
    int i = blockIdx.x * blockDim.x + threadIdx.x;   // 6.4M threads
    agg[i] = 0.0f;                                   // grid sized exactly
    if (i < N_NODES) deg[i] = 0.0f;
}

// ---------------------------------------------------------------------------
// Kernel 2: degree of destination nodes (scatter-add of ones over row).
// ---------------------------------------------------------------------------
__global__ void k_deg(const int* __restrict__ ei, float* __restrict__ deg) {
    int e = blockIdx.x * blockDim.x + threadIdx.x;
    if (e < N_EDGES) atomicAdd(&deg[ei[e]], 1.0f);   // ei[0:E] = row
}

// ---------------------------------------------------------------------------
// Kernel 3: deg_inv_sqrt = deg > 0 ? deg^-0.5 : 0
// ---------------------------------------------------------------------------
__global__ void k_dis(const float* __restrict__ deg, float* __restrict__ dis) {
    int n = blockIdx.x * blockDim.x + threadIdx.x;
    if (n < N_NODES) {
        float d = deg[n];
        dis[n] = (d > 0.0f) ? rsqrtf(d) : 0.0f;
    }
}

// ---------------------------------------------------------------------------
// Kernel 4: edge scatter. One wave32 per edge: each lane loads float4 of
// x[col], scales by norm, and does 4 contiguous global f32 atomics into
// agg[row]. x (25.6MB) and agg (25.6MB) both live in the 192MB L2.
// ---------------------------------------------------------------------------
__global__ void k_scatter(const float* __restrict__ x,
                          const int* __restrict__ ei,
                          const float* __restrict__ dis,
                          float* __restrict__ agg) {
    int gid  = blockIdx.x * blockDim.x + threadIdx.x;
    int edge = gid >> 5;                 // wave32: one edge per wave
    int lane = threadIdx.x & 31;
    if (edge >= N_EDGES) return;

    int r = ei[edge];                    // destination (row)
    int c = ei[N_EDGES + edge];          // source (col)
    float w = dis[r] * dis[c];

    const float4* xr = (const float4*)(x + (size_t)c * CH);
    float4 v = xr[lane];                 // GLOBAL_LOAD_B128, coalesced

    float* o = agg + (size_t)r * CH + lane * 4;
    atomicAdd(o + 0, v.x * w);           // global_atomic_add_f32 x4
    atomicAdd(o + 1, v.y * w);
    atomicAdd(o + 2, v.z * w);
    atomicAdd(o + 3, v.w * w);
}

// ---------------------------------------------------------------------------
// Kernel 5: out = agg @ W^T + b, in place (agg lives in d_out).
// One wave per 16-row block computes the full 16x128 output block with
// V_WMMA_F32_16X16X4_F32: 8 column tiles x (128/4) K-steps. All reads of the
// wave's 16 agg rows complete before its stores -> safe in-place.
// W is staged in LDS (64KB; WGP has 320KB).
// ---------------------------------------------------------------------------
__global__ void k_gemm(float* __restrict__ io,           // agg in / out
                       const float* __restrict__ W,      // [128,128] row-major
                       const float* __restrict__ bias) { // [128]
    __shared__ float Wl[CH * CH];                        // 64 KB

    int tid = threadIdx.x;
    // cooperative W preload: 4096 float4 / 256 threads = 16 each
    const float4* Wv  = (const float4*)W;
    float4*       Wlv = (float4*)Wl;
#pragma unroll
    for (int i = 0; i < 16; ++i) Wlv[tid + i * 256] = Wv[tid + i * 256];
    __syncthreads();

    int wave = tid >> 5;
    int lane = tid & 31;
    int rb = blockIdx.x * 8 + wave;      // row block (wave-uniform)
    if (rb >= ROW_BLOCKS) return;        // uniform exit: EXEC stays all-ones

    int m    = lane & 15;                // A-matrix row for this lane
    int kh   = (lane >> 4) * 2;          // K sub-offset: 0 (lanes 0-15) / 2
    int ncol = lane & 15;                // B/C/D column for this lane

    const float* arow = io + ((size_t)rb * 16 + m) * CH;

    v8f acc[8] = {};                     // 8 x (16x16 f32) accumulator tiles

    for (int k0 = 0; k0 < CH; k0 += 4) {
        v2f a;
        a.x = arow[k0 + kh];             // A[m][k0+kh],A[m][k0+kh+1]
        a.y = arow[k0 + kh + 1];
#pragma unroll
        for (int t = 0; t < 8; ++t) {
            const float* wrow = Wl + (t * 16 + ncol) * CH + k0 + kh;
            v2f b;
            b.x = wrow[0];               // B[k][n] = W[n][k]
            b.y = wrow[1];
            acc[t] = __builtin_amdgcn_wmma_f32_16x16x4_f32(
                false, a, false, b, (short)0, acc[t], false, false);
        }
    }

    // C/D layout: VGPR r -> row r (lanes 0-15) / row r+8 (lanes 16-31)
    int base_m = rb * 16 + (lane >> 4) * 8;
#pragma unroll
    for (int t = 0; t < 8; ++t) {
        int col = t * 16 + ncol;
        float bv = bias[col];
#pragma unroll
        for (int r = 0; r < 8; ++r) {
            io[(size_t)(base_m + r) * CH + col] = acc[t][r] + bv;
        }
    }
}

// ---------------------------------------------------------------------------
extern "C" void kernel_launch(void* const* d_in, const int* in_sizes, int n_in,
                              void* d_out, int out_size, void* d_ws, size_t ws_size,
                              hipStream_t stream) {
    const float* x  = (const float*)d_in[0];   // [50000,128]
    const int*   ei = (const int*)d_in[1];     // [2,625000]
    const float* W  = (const float*)d_in[2];   // [128,128]
    const float* b  = (const float*)d_in[3];   // [128]
    float* out = (float*)d_out;                // [50000,128]

    float* deg = (float*)d_ws;                 // 50000 f32
    float* dis = deg + N_NODES;                // 50000 f32  (400KB total ws)

    // 1) zero aggregate (in d_out) + deg
    k_zero<<<(N_NODES * CH) / 256, 256, 0, stream>>>(out, deg);
    // 2) degrees
    k_deg<<<(N_EDGES + 255) / 256, 256, 0, stream>>>(ei, deg);
    // 3) deg^-1/2
    k_dis<<<(N_NODES + 255) / 256, 256, 0, stream>>>(deg, dis);
    // 4) gather-scale-scatter (one wave per edge)
    k_scatter<<<(N_EDGES * 32) / 256, 256, 0, stream>>>(x, ei, dis, out);
    // 5) WMMA GEMM + bias, in place
    k_gemm<<<(ROW_BLOCKS + 7) / 8, 256, 0, stream>>>(out, W, b);
}